// TopkReLU_47278999994879
// MI455X (gfx1250) — compile-verified
//
#include <hip/hip_runtime.h>
#include <stdint.h>

// ---------------------------------------------------------------------------
// TopK-ReLU: per row (8192 f32), zero everything below the 1024th-largest.
// One workgroup per row; row staged in LDS once via async global->LDS b128
// (CDNA5 ASYNCcnt path); 4-pass MSB radix select over LDS with wave32
// shuffle suffix-scans (3 barriers/pass); single non-temporal store pass.
// HBM traffic = read-once + write-once = 512 MB total -> ~22 us at 23.3 TB/s.
// ---------------------------------------------------------------------------

typedef float    __attribute__((ext_vector_type(4))) f32x4;
typedef uint32_t __attribute__((ext_vector_type(4))) u32x4;

#define ROW      8192
#define KSEL     (ROW / 8)        // 1024
#define THREADS  256              // 8 wave32
#define NWAVE    (THREADS / 32)
#define CHUNKS   (ROW / (THREADS * 4))   // 8 b128 ops per thread
#define HSTRIDE  264              // per-wave hist stride; 264 % 64 == 8 -> each
                                  // wave's copy of a bin sits in a distinct bank

// Order-preserving map: larger float  <->  larger uint32.
__device__ __forceinline__ uint32_t fmap(uint32_t b) {
  return (b & 0x80000000u) ? ~b : (b | 0x80000000u);
}

__global__ __launch_bounds__(THREADS)
void topk_relu_kernel(const float* __restrict__ x, float* __restrict__ out, int rows) {
  __shared__ __align__(16) uint32_t rowbits[ROW];       // 32 KB: raw f32 bits of the row
  __shared__ uint32_t whist[NWAVE * HSTRIDE];           // 8 private 256-bin histograms
  __shared__ uint32_t wtot[NWAVE];                      // per-wave scan totals
  __shared__ uint32_t s_digit, s_gt;

  const int row = blockIdx.x;
  if (row >= rows) return;
  const int tid  = threadIdx.x;
  const int lane = tid & 31;
  const int wid  = tid >> 5;
  const float* __restrict__ xrow = x + (size_t)row * ROW;
  float* __restrict__ orow       = out + (size_t)row * ROW;

  // ---- Stage row into LDS: async global->LDS b128 (tracked by ASYNCcnt) ----
  // vDst VGPR carries the LDS byte offset (low 32 bits of the flat shared
  // address; aperture sits in the high half), vAddr the 64-bit global address.
  #pragma unroll
  for (int j = 0; j < CHUNKS; ++j) {
    const int i = (tid + j * THREADS) * 4;
    const uint32_t lds_off = (uint32_t)(uintptr_t)(&rowbits[i]);
    const uint64_t gaddr   = (uint64_t)(uintptr_t)(xrow + i);
    asm volatile("global_load_async_to_lds_b128 %0, %1, off"
                 :
                 : "v"(lds_off), "v"(gaddr)
                 : "memory");
  }
  asm volatile("s_wait_asynccnt 0" ::: "memory");
  __syncthreads();

  // ---- 4-pass MSB-first radix select for the KSEL-th largest value ----
  uint32_t prefix = 0;      // digits determined so far, placed at their bit positions
  uint32_t kk     = KSEL;   // rank within the current candidate set

  #pragma unroll
  for (int pass = 0; pass < 4; ++pass) {
    const int      shift  = 8 * (3 - pass);
    const uint32_t himask = (pass == 0) ? 0u : (0xFFFFFFFFu << (8 * (4 - pass)));

    // zero this wave's private histogram (own-wave only: no barrier needed
    // before the own-wave atomics below; LDS ops are in-order per wave)
    #pragma unroll
    for (int b = 0; b < 256 / 32; ++b) whist[wid * HSTRIDE + lane + b * 32] = 0u;

    #pragma unroll
    for (int j = 0; j < CHUNKS; ++j) {
      const int i = (tid + j * THREADS) * 4;
      u32x4 q = *(const u32x4*)(&rowbits[i]);
      #pragma unroll
      for (int c = 0; c < 4; ++c) {
        const uint32_t key   = fmap(q[c]);
        const bool     ok    = ((key ^ prefix) & himask) == 0u;
        const uint32_t digit = (key >> shift) & 0xFFu;
        if (pass == 0) {
          // Pass 0 digits = sign+exponent of N(0,1): heavily clustered.
          // Wave32 ballot-match aggregates equal digits; leader lane adds.
          const uint32_t valid = (uint32_t)__ballot(ok);
          uint32_t m = valid;
          #pragma unroll
          for (int b = 0; b < 8; ++b) {
            const uint32_t vote = (uint32_t)__ballot((digit >> b) & 1u);
            m &= ((digit >> b) & 1u) ? vote : ~vote;
          }
          if (ok && lane == (__ffs(m) - 1)) {
            atomicAdd(&whist[wid * HSTRIDE + digit], (uint32_t)__popc(m));
          }
        } else {
          // Later passes: candidates are sparse and digits near-uniform.
          if (ok) atomicAdd(&whist[wid * HSTRIDE + digit], 1u);
        }
      }
    }
    __syncthreads();   // (1) all waves' histograms complete

    // Thread t owns bin t: reduce the 8 private histograms, then a wave32
    // shuffle suffix-scan (bins t..wave-end), then cross-wave fixup via LDS.
    uint32_t v = 0;
    #pragma unroll
    for (int w = 0; w < NWAVE; ++w) v += whist[w * HSTRIDE + tid];
    #pragma unroll
    for (int off = 1; off < 32; off <<= 1) {
      const uint32_t t = __shfl(v, lane + off);
      v += (lane + off < 32) ? t : 0u;
    }
    if (lane == 0) wtot[wid] = v;   // wave's 32-bin total (suffix at its lowest bin)
    __syncthreads();   // (2) wtot visible
    uint32_t add = 0;
    #pragma unroll
    for (int w = 0; w < NWAVE; ++w) add += (w > wid) ? wtot[w] : 0u;
    v += add;                                  // v = #candidates with digit >= tid
    const uint32_t nv = __shfl(v, lane + 1);
    const uint32_t gt = (lane == 31) ? add : nv;  // #candidates with digit > tid
    // kk-th largest lands in bin d where v >= kk > gt; exactly one thread hits.
    if (v >= kk && gt < kk) { s_digit = (uint32_t)tid; s_gt = gt; }
    __syncthreads();   // (3) selection broadcast
    prefix |= s_digit << shift;
    kk     -= s_gt;
  }

  // Decode the mapped key of the KSEL-th largest back to a float threshold.
  const uint32_t tb = (prefix & 0x80000000u) ? (prefix & 0x7FFFFFFFu) : ~prefix;
  const float    thr = __uint_as_float(tb);

  // ---- Apply threshold from LDS; non-temporal b128 stores (write-once) ----
  #pragma unroll
  for (int j = 0; j < CHUNKS; ++j) {
    const int i = (tid + j * THREADS) * 4;
    u32x4 q = *(const u32x4*)(&rowbits[i]);
    f32x4 o;
    #pragma unroll
    for (int c = 0; c < 4; ++c) {
      const float xv = __uint_as_float(q[c]);
      o[c] = (xv >= thr) ? xv : 0.0f;
    }
    __builtin_nontemporal_store(o, (f32x4*)(orow + i));
  }
}

extern "C" void kernel_launch(void* const* d_in, const int* in_sizes, int n_in,
                              void* d_out, int out_size, void* d_ws, size_t ws_size,
                              hipStream_t stream) {
  const float* x = (const float*)d_in[0];
  float* out = (float*)d_out;
  const int rows = in_sizes[0] / ROW;   // 8192 for the reference shapes
  topk_relu_kernel<<<dim3(rows), dim3(THREADS), 0, stream>>>(x, out, rows);
}